// AttnResidual_26645977105192
// MI455X (gfx1250) — compile-verified
//
#include <hip/hip_runtime.h>

// Problem constants (from the reference)
#define N_ROWS   8            // 7 completed blocks + 1 partial block
#define N_BLK    7
#define B_DIM    2
#define L_DIM    4096
#define D_DIM    2048
#define THREADS  256
#define WAVES    (THREADS / 32)

// ---------------------------------------------------------------------------
// CDNA5 async global->LDS copy (ASYNCcnt-tracked) + staged waits.
// Inline asm per CDNA5 ISA / CDNA5 ISA (GLOBAL_LOAD_ASYNC_
// TO_LDS_B128, opcode 98): operands are vdst(LDS byte offset), vaddr(64-bit
// global address), saddr(off). Inline asm is portable across both toolchains
// and avoids the builtin's uncharacterized signature.
// ---------------------------------------------------------------------------
typedef __attribute__((address_space(3))) float lds_float;

__device__ __forceinline__ void async_copy_f4_to_lds(float* lds_generic,
                                                     const float* gsrc) {
  lds_float* l = (lds_float*)lds_generic;   // generic -> LDS segment offset (32-bit)
  asm volatile("global_load_async_to_lds_b128 %0, %1, off"
               :: "v"(l), "v"(gsrc)
               : "memory");
}

template <int CNT>
__device__ __forceinline__ void wait_asynccnt_le() {
#if __has_builtin(__builtin_amdgcn_s_wait_asynccnt)
  __builtin_amdgcn_s_wait_asynccnt(CNT);
#else
  asm volatile("s_wait_asynccnt %0" :: "n"(CNT) : "memory");
#endif
}

// ---------------------------------------------------------------------------
// One workgroup per (b,l): stream the 8 x 2048 f32 tile HBM -> LDS with the
// async engine, overlap per-row ssq / proj-dot reductions with the in-flight
// DMA, softmax over the N=8 axis, then weighted-sum the raw rows to d_out.
// ---------------------------------------------------------------------------
__global__ __launch_bounds__(THREADS)
void AttnResidual_gfx1250_kernel(const float* __restrict__ blocks,
                                 const float* __restrict__ partial,
                                 const float* __restrict__ norm_scale,
                                 const float* __restrict__ proj,
                                 float* __restrict__ out) {
  __shared__ float rows[N_ROWS * D_DIM];   // 64 KB staging tile
  __shared__ float red[WAVES][16];         // cross-wave reduction scratch
  __shared__ float lgw[16];                // [0..7]=logits, [8..15]=weights

  const int  tid = threadIdx.x;
  const long bl  = blockIdx.x;             // flat (b*L + l), 0..8191
  const long rowOff = bl * (long)D_DIM;    // element offset of (b,l,:) in [B,L,D]

  const int c0 = tid;                      // this lane's float4 chunk ids in a row
  const int c1 = tid + THREADS;            // (2048 floats = 512 float4 chunks)

  // ---- Issue all 16 async b128 copies (2 per row, rows 0..7 in order) ----
#pragma unroll
  for (int n = 0; n < N_ROWS; ++n) {
    const float* src = (n < N_BLK)
        ? (blocks + (long)n * (long)(B_DIM * L_DIM) * (long)D_DIM + rowOff)
        : (partial + rowOff);
    async_copy_f4_to_lds(&rows[n * D_DIM + 4 * c0], src + 4 * c0);
    async_copy_f4_to_lds(&rows[n * D_DIM + 4 * c1], src + 4 * c1);
  }

  // ---- Fold norm_scale into proj for this lane's 8 d-indices (L2-resident) --
  const float4* pj = (const float4*)proj;
  const float4* ns = (const float4*)norm_scale;
  float4 pa = pj[c0], pb = pj[c1];
  float4 sa = ns[c0], sb = ns[c1];
  pa.x *= sa.x; pa.y *= sa.y; pa.z *= sa.z; pa.w *= sa.w;
  pb.x *= sb.x; pb.y *= sb.y; pb.z *= sb.z; pb.w *= sb.w;

  // ---- Pass 1: consume rows as the DMA lands them (in-order completion) ----
  float ssq[N_ROWS], dot[N_ROWS];
#define PROC_ROW(n, cnt)                                                      \
  {                                                                           \
    wait_asynccnt_le<cnt>();                                                  \
    const float4* r = (const float4*)&rows[(n) * D_DIM];                      \
    float4 a = r[c0], b = r[c1];                                              \
    ssq[n] = a.x * a.x + a.y * a.y + a.z * a.z + a.w * a.w +                  \
             b.x * b.x + b.y * b.y + b.z * b.z + b.w * b.w;                   \
    dot[n] = pa.x * a.x + pa.y * a.y + pa.z * a.z + pa.w * a.w +              \
             pb.x * b.x + pb.y * b.y + pb.z * b.z + pb.w * b.w;               \
  }
  PROC_ROW(0, 14) PROC_ROW(1, 12) PROC_ROW(2, 10) PROC_ROW(3, 8)
  PROC_ROW(4, 6)  PROC_ROW(5, 4)  PROC_ROW(6, 2)  PROC_ROW(7, 0)
#undef PROC_ROW

  // ---- Reduce: wave32 butterfly, then across the 8 waves via LDS ----
#pragma unroll
  for (int k = 0; k < N_ROWS; ++k) {
#pragma unroll
    for (int m = 16; m >= 1; m >>= 1) {
      ssq[k] += __shfl_xor(ssq[k], m, 32);
      dot[k] += __shfl_xor(dot[k], m, 32);
    }
  }
  const int lane = tid & 31;
  const int wv   = tid >> 5;
  if (lane == 0) {
#pragma unroll
    for (int k = 0; k < N_ROWS; ++k) {
      red[wv][k]          = ssq[k];
      red[wv][N_ROWS + k] = dot[k];
    }
  }
  __syncthreads();

  // ---- Logits: logit_n = (p . V_n) / sqrt(mean(V_n^2) + 1e-6) ----
  if (tid < N_ROWS) {
    float s = 0.f, d = 0.f;
#pragma unroll
    for (int w = 0; w < WAVES; ++w) {
      s += red[w][tid];
      d += red[w][N_ROWS + tid];
    }
    lgw[tid] = d / sqrtf(s * (1.0f / (float)D_DIM) + 1e-6f);
  }
  __syncthreads();

  // ---- Softmax over the N=8 axis (redundantly in 8 threads) ----
  if (tid < N_ROWS) {
    float m = lgw[0];
#pragma unroll
    for (int k = 1; k < N_ROWS; ++k) m = fmaxf(m, lgw[k]);
    float denom = 0.f;
#pragma unroll
    for (int k = 0; k < N_ROWS; ++k) denom += expf(lgw[k] - m);
    lgw[N_ROWS + tid] = expf(lgw[tid] - m) / denom;
  }
  __syncthreads();

  // ---- Pass 2: out[d] = sum_n w_n * V[n,d] from the LDS tile ----
  float w[N_ROWS];
#pragma unroll
  for (int k = 0; k < N_ROWS; ++k) w[k] = lgw[N_ROWS + k];

  float4 a0 = {0.f, 0.f, 0.f, 0.f};
  float4 a1 = {0.f, 0.f, 0.f, 0.f};
#pragma unroll
  for (int n = 0; n < N_ROWS; ++n) {
    const float4* r = (const float4*)&rows[n * D_DIM];
    float4 x = r[c0], y = r[c1];
    a0.x += w[n] * x.x; a0.y += w[n] * x.y; a0.z += w[n] * x.z; a0.w += w[n] * x.w;
    a1.x += w[n] * y.x; a1.y += w[n] * y.y; a1.z += w[n] * y.z; a1.w += w[n] * y.w;
  }
  float4* o = (float4*)(out + rowOff);
  o[c0] = a0;
  o[c1] = a1;
}

// ---------------------------------------------------------------------------
extern "C" void kernel_launch(void* const* d_in, const int* in_sizes, int n_in,
                              void* d_out, int out_size, void* d_ws,
                              size_t ws_size, hipStream_t stream) {
  (void)in_sizes; (void)n_in; (void)out_size; (void)d_ws; (void)ws_size;
  const float* blocks     = (const float*)d_in[0];  // [7,2,4096,2048] f32
  const float* partial    = (const float*)d_in[1];  // [2,4096,2048]   f32
  const float* norm_scale = (const float*)d_in[2];  // [2048]          f32
  const float* proj       = (const float*)d_in[3];  // [2048]          f32
  float* out              = (float*)d_out;          // [2,4096,2048]   f32

  dim3 grid(B_DIM * L_DIM);  // 8192 workgroups, one per (b,l)
  AttnResidual_gfx1250_kernel<<<grid, THREADS, 0, stream>>>(
      blocks, partial, norm_scale, proj, out);
}